// MemoryMatrixBlock_20976620274028
// MI455X (gfx1250) — compile-verified
//
#include <hip/hip_runtime.h>
#include <cstdint>

// ---------------------------------------------------------------------------
// MemoryMatrixBlock on MI455X (gfx1250, wave32).
// HBM-bound (604 MB fp32 memory matrices vs ~4.8 GFLOP -> ~7.5 FLOP/B), so
// fp32 WMMA (V_WMMA_F32_16X16X4_F32) handles both GEMMs; launch order keeps
// the 33.5MB mem[m] tile warm in the 192MB L2 between the att pass and the
// read-back pass. Padding rows of the 16-row WMMA A-operand read from a
// zeroed scratch row via a hoisted pointer select (no per-iteration exec
// branching in the inner loops).
// ---------------------------------------------------------------------------

#define B_ 8
#define C_ 256
#define H_ 48
#define GRID_ 3
#define M_ 9            // number of windows
#define WS_ 16          // window size
#define S_ 128          // slots
#define D_ 65536        // WS*WS*C
#define K_ 5            // top-k threshold
#define EPS_ 1e-12f

typedef __attribute__((ext_vector_type(2))) float v2f;
typedef __attribute__((ext_vector_type(8))) float v8f;

// ---------------------------------------------------------------------------
// K1: window_partition  x(B,C,H,H) -> q(B,M,D), d = (i*16+j)*256 + c
// ---------------------------------------------------------------------------
__global__ void pack_q_kernel(const float* __restrict__ x, float* __restrict__ q) {
    size_t tid = (size_t)blockIdx.x * blockDim.x + threadIdx.x;   // over B*M*D
    int d  = (int)(tid % D_);
    size_t bm = tid / D_;
    int m  = (int)(bm % M_);
    int b  = (int)(bm / M_);
    int c  = d % C_;
    int r  = d / C_;
    int i  = r >> 4;          // row in window
    int j  = r & 15;          // col in window
    int gh = m / GRID_, gw = m % GRID_;
    int h  = gh * WS_ + i, w = gw * WS_ + j;
    q[tid] = x[(((size_t)b * C_ + c) * H_ + h) * H_ + w];
}

// ---------------------------------------------------------------------------
// K2: att[memIdx][b][m][s] += q[b,m,:] . mem[m][s,:]   (split-K over D)
// One block = (memIdx, m, k-chunk); 8 waves, wave w owns s-tile [16w,16w+16).
// WMMA f32 16x16x4: A = q rows (b, rows 8..15 read a zeroed scratch row).
// ---------------------------------------------------------------------------
#define NCHUNK_ 64
#define CHUNK_ (D_ / NCHUNK_)   // 1024

__global__ void att_gemm_kernel(const float* __restrict__ q,
                                const float* __restrict__ mem1,
                                const float* __restrict__ mem2,
                                const float* __restrict__ zrow,
                                float* __restrict__ att) {
    int bid    = blockIdx.x;                 // (memIdx*M_ + m)*NCHUNK_ + chunk
    int chunk  = bid % NCHUNK_;
    int mm     = bid / NCHUNK_;
    int memIdx = mm / M_;
    int m      = mm % M_;
    const float* mem = memIdx ? mem2 : mem1;

    int lane  = threadIdx.x & 31;
    int wid   = threadIdx.x >> 5;            // 0..7 -> s-tile
    int s0    = wid * 16;
    int n     = lane & 15;                   // A: M row / B: N col
    int kbase = (lane >> 4) * 2;             // K pair per lane half

    // Hoisted guard: padding rows read from a zero-filled scratch row.
    const float* qrow = (n < 8) ? (q + ((size_t)n * M_ + m) * D_) : zrow;
    const float* brow = mem + ((size_t)m * S_ + (s0 + n)) * D_;

    int d0 = chunk * CHUNK_;
    v8f acc0 = {}, acc1 = {};
    for (int d = d0; d < d0 + CHUNK_; d += 8) {
        v2f a0 = *(const v2f*)(qrow + d + kbase);
        v2f b0 = *(const v2f*)(brow + d + kbase);
        v2f a1 = *(const v2f*)(qrow + d + 4 + kbase);
        v2f b1 = *(const v2f*)(brow + d + 4 + kbase);
        __builtin_prefetch(brow + d + 256, 0, 1);   // global_prefetch_b8
        acc0 = __builtin_amdgcn_wmma_f32_16x16x4_f32(
            false, a0, false, b0, (short)0, acc0, false, false);
        acc1 = __builtin_amdgcn_wmma_f32_16x16x4_f32(
            false, a1, false, b1, (short)0, acc1, false, false);
    }
    v8f acc = acc0 + acc1;

    // D layout: lanes 0..15 hold (M=r, N=lane) in VGPR r -> valid b rows 0..7
    if (lane < 16) {
        int s = s0 + lane;
#pragma unroll
        for (int r = 0; r < 8; ++r)
            atomicAdd(att + (((size_t)memIdx * B_ + r) * M_ + m) * S_ + s, acc[r]);
    }
}

// ---------------------------------------------------------------------------
// K3: softmax over S, top-5 threshold, hard-shrink-relu, L1 renorm.
// One wave32 per (memIdx, b, m) row; each lane owns 4 of 128 slots.
// ---------------------------------------------------------------------------
__global__ void softmax_topk_kernel(const float* __restrict__ att,
                                    float* __restrict__ yw) {
    int bi     = blockIdx.x;                 // 0..143
    int memIdx = bi / (B_ * M_);
    int rem    = bi % (B_ * M_);
    int b      = rem / M_;
    int m      = rem % M_;
    size_t base = (((size_t)memIdx * B_ + b) * M_ + m) * S_;
    int lane = threadIdx.x;

    float v[4];
#pragma unroll
    for (int t = 0; t < 4; ++t) v[t] = att[base + lane + 32 * t];

    // softmax
    float mx = fmaxf(fmaxf(v[0], v[1]), fmaxf(v[2], v[3]));
    for (int o = 16; o > 0; o >>= 1) mx = fmaxf(mx, __shfl_xor(mx, o, 32));
    float e[4], sum = 0.0f;
#pragma unroll
    for (int t = 0; t < 4; ++t) { e[t] = __expf(v[t] - mx); sum += e[t]; }
    for (int o = 16; o > 0; o >>= 1) sum += __shfl_xor(sum, o, 32);
    float inv = 1.0f / sum;
    float y[4];
#pragma unroll
    for (int t = 0; t < 4; ++t) y[t] = e[t] * inv;

    // 5th-largest value as threshold: 5 rounds of max + remove one instance
    bool removed[4] = {false, false, false, false};
    float thres = 0.0f;
    for (int it = 0; it < K_; ++it) {
        float lb = -1.0f; int li = -1;
#pragma unroll
        for (int t = 0; t < 4; ++t)
            if (!removed[t] && y[t] > lb) { lb = y[t]; li = t; }
        float gb = lb;
        for (int o = 16; o > 0; o >>= 1) gb = fmaxf(gb, __shfl_xor(gb, o, 32));
        unsigned long long msk = __ballot(lb == gb && li >= 0);
        int first = __ffsll(msk) - 1;
        if (lane == first) removed[li] = true;
        thres = gb;
    }

    // hard shrink relu + L1 normalize
    float yh[4], s1 = 0.0f;
#pragma unroll
    for (int t = 0; t < 4; ++t) {
        float d  = y[t] - thres;
        float rl = d > 0.0f ? d : 0.0f;
        yh[t] = rl * y[t] / (fabsf(d) + EPS_);
        s1 += fabsf(yh[t]);
    }
    for (int o = 16; o > 0; o >>= 1) s1 += __shfl_xor(s1, o, 32);
    float nrm = 1.0f / fmaxf(s1, EPS_);
#pragma unroll
    for (int t = 0; t < 4; ++t) yw[base + lane + 32 * t] = yh[t] * nrm;
}

// ---------------------------------------------------------------------------
// K4: out[b,m,d] = sum_s y1*mem1 + sum_s y2*mem2  (K=S=128; both memories
// accumulate into the same WMMA accumulators -> fused sum).
// Block = 128 threads (4 waves); wave owns a 16-wide d tile; grid over (m, d).
// ---------------------------------------------------------------------------
#define DBLK_ 64

__global__ void out_gemm_kernel(const float* __restrict__ yw,
                                const float* __restrict__ mem1,
                                const float* __restrict__ mem2,
                                const float* __restrict__ zrow,
                                float* __restrict__ outb) {
    int nblkPerM = D_ / DBLK_;               // 1024
    int m    = blockIdx.x / nblkPerM;
    int dblk = blockIdx.x % nblkPerM;
    int lane = threadIdx.x & 31;
    int wid  = threadIdx.x >> 5;
    int dtile = dblk * DBLK_ + wid * 16;
    int n     = lane & 15;
    int kbase = (lane >> 4) * 2;

    v8f acc0 = {}, acc1 = {};
    for (int memIdx = 0; memIdx < 2; ++memIdx) {
        const float* mem  = memIdx ? mem2 : mem1;
        const float* yrow = (n < 8)
            ? (yw + (((size_t)memIdx * B_ + n) * M_ + m) * S_) : zrow;
        const float* mb   = mem + (size_t)m * S_ * D_ + dtile + n;
        for (int k0 = 0; k0 < S_; k0 += 8) {
            v2f a0 = *(const v2f*)(yrow + k0 + kbase);
            v2f a1 = *(const v2f*)(yrow + k0 + 4 + kbase);
            v2f b0, b1;
            b0.x = mb[(size_t)(k0 + kbase)     * D_];
            b0.y = mb[(size_t)(k0 + kbase + 1) * D_];
            b1.x = mb[(size_t)(k0 + 4 + kbase)     * D_];
            b1.y = mb[(size_t)(k0 + 4 + kbase + 1) * D_];
            if (k0 + 16 < S_)
                __builtin_prefetch(mb + (size_t)(k0 + 16 + kbase) * D_, 0, 1);
            acc0 = __builtin_amdgcn_wmma_f32_16x16x4_f32(
                false, a0, false, b0, (short)0, acc0, false, false);
            acc1 = __builtin_amdgcn_wmma_f32_16x16x4_f32(
                false, a1, false, b1, (short)0, acc1, false, false);
        }
    }
    v8f acc = acc0 + acc1;
    if (lane < 16) {
#pragma unroll
        for (int r = 0; r < 8; ++r)         // b = r, d = dtile + lane
            outb[((size_t)r * M_ + m) * D_ + dtile + lane] = acc[r];
    }
}

// ---------------------------------------------------------------------------
// K5: window_reverse  outb(B,M,D) -> out(B,C,H,H)   (coalesced writes)
// ---------------------------------------------------------------------------
__global__ void unpack_out_kernel(const float* __restrict__ outb,
                                  float* __restrict__ out) {
    size_t tid = (size_t)blockIdx.x * blockDim.x + threadIdx.x;   // over B*C*H*H
    int w = (int)(tid % H_);
    size_t t = tid / H_;
    int h = (int)(t % H_); t /= H_;
    int c = (int)(t % C_);
    int b = (int)(t / C_);
    int gh = h / WS_, gw = w / WS_, i = h % WS_, j = w % WS_;
    int m = gh * GRID_ + gw;
    int d = (i * WS_ + j) * C_ + c;
    out[tid] = outb[((size_t)b * M_ + m) * D_ + d];
}

// ---------------------------------------------------------------------------
extern "C" void kernel_launch(void* const* d_in, const int* in_sizes, int n_in,
                              void* d_out, int out_size, void* d_ws, size_t ws_size,
                              hipStream_t stream) {
    const float* x    = (const float*)d_in[0];
    const float* mem1 = (const float*)d_in[1];
    const float* mem2 = (const float*)d_in[2];
    float* out = (float*)d_out;

    float* wsf = (float*)d_ws;
    const size_t QN  = (size_t)B_ * M_ * D_;         // 4,718,592 floats
    const size_t AN  = (size_t)2 * B_ * M_ * S_;     // 18,432 floats
    float* q    = wsf;                                // 18.9 MB
    float* outb = wsf + QN;                           // 18.9 MB
    float* att  = wsf + 2 * QN;                       // AN floats (atomics)
    float* zrow = att + AN;                           // D_ floats, stays zero
    float* yw   = zrow + D_;                          // AN floats

    // att accumulated with atomics + zrow must be zero -> clear both
    // (contiguous region) every call; graph-capture-safe.
    hipMemsetAsync(att, 0, (AN + (size_t)D_) * sizeof(float), stream);

    pack_q_kernel<<<(int)(QN / 256), 256, 0, stream>>>(x, q);
    att_gemm_kernel<<<2 * M_ * NCHUNK_, 256, 0, stream>>>(q, mem1, mem2, zrow, att);
    softmax_topk_kernel<<<2 * B_ * M_, 32, 0, stream>>>(att, yw);
    out_gemm_kernel<<<M_ * (D_ / DBLK_), 128, 0, stream>>>(yw, mem1, mem2, zrow, outb);
    unpack_out_kernel<<<(int)(QN / 256), 256, 0, stream>>>(outb, out);
}